// HGNNLayer_49271864820040
// MI455X (gfx1250) — compile-verified
//
#include <hip/hip_runtime.h>
#include <hip/hip_bf16.h>
#include <math.h>

// ---------------------------------------------------------------------------
// HGNN layer for MI455X (gfx1250, wave32).
// Memory-bound (~16 GFLOP vs ~1 GB traffic) -> full fp32 WMMA
// (V_WMMA_F32_16X16X4_F32) for all GEMMs; TDM tensor_load_to_lds stages the
// per-block A tiles (with hardware LDS padding to avoid bank conflicts);
// scatter phases use L2 f32 atomics.
// ---------------------------------------------------------------------------

#define NN   50000
#define EE   200000
#define FN   128
#define FE   64
#define NH   4
#define DD   64
#define HD   256   /* NH * DD */

// LDS row padding (4 dwords) so GEMM-A column reads hit 16 distinct banks
#define PAD  4

typedef float v2f __attribute__((ext_vector_type(2)));
typedef float v8f __attribute__((ext_vector_type(8)));
typedef unsigned u32x4 __attribute__((ext_vector_type(4)));
typedef int i32x4 __attribute__((ext_vector_type(4)));
typedef int i32x8 __attribute__((ext_vector_type(8)));

#if __has_builtin(__builtin_amdgcn_wmma_f32_16x16x4_f32)
#define HAS_WMMA_F32 1
#else
#define HAS_WMMA_F32 0
#endif

// TDM used unconditionally: builtin is present on both probed toolchains
// (ROCm 7.2 clang-22: 5 args; amdgpu-toolchain clang-23: 6 args). A successful
// compile of this file therefore proves tensor_load_to_lds is emitted.
#define HAS_TDM 1

__device__ inline void wait_tensorcnt0() {
#if __has_builtin(__builtin_amdgcn_s_wait_tensorcnt)
    __builtin_amdgcn_s_wait_tensorcnt(0);
#else
    asm volatile("s_wait_tensorcnt 0x0" ::: "memory");
#endif
}

// 2D TDM load: rows x rowlen fp32 elements, row stride in elements, with
// hardware LDS padding of PAD dwords inserted after every `rowlen` dwords.
// pad_interval code: dwords-before-pad = 2^(code+1); pad_amount code: n-1.
// Descriptor packing per CDNA5 ISA.
__device__ inline void tdm_load_2d(unsigned lds_addr, const float* gptr,
                                   unsigned rows, unsigned rowlen,
                                   unsigned row_stride, unsigned pad_code) {
    unsigned long long ga = (unsigned long long)(uintptr_t)gptr;
    u32x4 g0;
    g0[0] = 1u;                                        // count=1 valid descriptor
    g0[1] = lds_addr;                                  // LDS byte address
    g0[2] = (unsigned)(ga & 0xFFFFFFFFu);              // global_addr[31:0]
    g0[3] = (unsigned)((ga >> 32) & 0x01FFFFFFu) | (2u << 30);  // hi bits | type=2
    i32x8 g1;
    g1[0] = (int)((2u << 16)                           // data_size = 4B
                | (1u << 20)                           // pad_enable
                | (pad_code << 22)                     // pad_interval
                | ((PAD - 1u) << 25));                 // pad_amount = 4 dwords
    g1[1] = (int)((rowlen & 0xFFFFu) << 16);           // tensor_dim0[15:0]
    g1[2] = (int)(((rowlen >> 16) & 0xFFFFu) | ((rows & 0xFFFFu) << 16)); // td0 hi|td1 lo
    g1[3] = (int)(((rows >> 16) & 0xFFFFu) | ((rowlen & 0xFFFFu) << 16)); // td1 hi|tile_dim0
    g1[4] = (int)(rows & 0xFFFFu);                     // tile_dim1 (tile_dim2=0)
    g1[5] = (int)row_stride;                           // tensor_dim0_stride lo32
    g1[6] = 0;
    g1[7] = 0;
    i32x4 z = {};
#if __clang_major__ >= 23
    i32x8 z8 = {};
    __builtin_amdgcn_tensor_load_to_lds(g0, g1, z, z, z8, 0);
#else
    __builtin_amdgcn_tensor_load_to_lds(g0, g1, z, z, 0);
#endif
}

// One 16x16 fp32 output tile per wave, accumulated over K (multiple of 4).
// A 16x4 f32 : lane<16 -> M=lane, K=k0,k0+1 ; lane>=16 -> K=k0+2,k0+3
// B 4x16 f32 : lane<16 -> N=lane, K=k0,k0+1 ; lane>=16 -> K=k0+2,k0+3
// C/D        : n = lane&15, rows m = (lane>>4)*8 + r
template <int K, class FA, class FB>
__device__ inline v8f gemm_tile_f32(FA loadA, FB loadB, v8f c) {
    const int lane = threadIdx.x & 31;
#if HAS_WMMA_F32
    const int kh = (lane >> 4) << 1;
    const int mn = lane & 15;
#pragma unroll
    for (int k0 = 0; k0 < K; k0 += 4) {
        v2f a, b;
        a.x = loadA(mn, k0 + kh);
        a.y = loadA(mn, k0 + kh + 1);
        b.x = loadB(k0 + kh, mn);
        b.y = loadB(k0 + kh + 1, mn);
        c = __builtin_amdgcn_wmma_f32_16x16x4_f32(false, a, false, b,
                                                  (short)0, c, false, false);
    }
#else
    const int n = lane & 15;
    const int mbase = (lane >> 4) * 8;
    for (int k = 0; k < K; ++k) {
        float bv = loadB(k, n);
#pragma unroll
        for (int r = 0; r < 8; ++r) c[r] += loadA(mbase + r, k) * bv;
    }
#endif
    return c;
}

__device__ inline unsigned f2ord(float f) {
    unsigned u = __float_as_uint(f);
    return (u & 0x80000000u) ? ~u : (u | 0x80000000u);
}
__device__ inline float ord2f(unsigned k) {
    unsigned u = (k & 0x80000000u) ? (k & 0x7FFFFFFFu) : ~k;
    return __uint_as_float(u);
}
__device__ inline float relu_f(float x) { return x > 0.0f ? x : 0.0f; }
__device__ inline float sigmoid_f(float x) { return 1.0f / (1.0f + __expf(-x)); }

// ---------------------------------------------------------------------------
// K1: feat_src = node_feats @ W_n2n^T + b_n2n          (N x 256)
// A tile (16 x 128) staged into LDS once per block via TDM.
// ---------------------------------------------------------------------------
__global__ void k_proj_nodes(const float* __restrict__ A,      // N x 128
                             const float* __restrict__ W,      // 256 x 128
                             const float* __restrict__ bias,   // 256
                             float* __restrict__ out) {        // N x 256
    __shared__ float sA[16][FN + PAD];
    const int row0 = blockIdx.x * 16;
    const int wid  = threadIdx.x >> 5;
    const int lane = threadIdx.x & 31;

    if (wid == 0) {
        tdm_load_2d((unsigned)(uintptr_t)&sA[0][0], A + (size_t)row0 * FN,
                    16u, (unsigned)FN, (unsigned)FN, 6u /* pad every 128 dw */);
        wait_tensorcnt0();
    }
    __syncthreads();

#pragma unroll
    for (int j = 0; j < 2; ++j) {
        const int col0 = (2 * wid + j) * 16;
        v8f c = {};
        c = gemm_tile_f32<FN>(
            [&](int m, int k) { return sA[m][k]; },
            [&](int k, int n) { return W[(size_t)(col0 + n) * FN + k]; }, c);
        const int n = lane & 15;
        const float bv = bias[col0 + n];
#pragma unroll
        for (int r = 0; r < 8; ++r) {
            int m = (lane >> 4) * 8 + r;
            out[(size_t)(row0 + m) * HD + col0 + n] = c[r] + bv;
        }
    }
}

// ---------------------------------------------------------------------------
// K2: feat_e = edge_feats @ W_e2n^T + b_e2n  (E x 256), fused with
//     new_edge_feats = relu(feat_e).mean(heads)  (E x 64)
// ---------------------------------------------------------------------------
__global__ void k_proj_edges(const float* __restrict__ A,      // E x 64
                             const float* __restrict__ W,      // 256 x 64
                             const float* __restrict__ bias,   // 256
                             float* __restrict__ feat_e,       // E x 256
                             float* __restrict__ out_e) {      // E x 64
    __shared__ float sAe[16][FE + PAD];
    __shared__ float sFE[16][HD];
    const int row0 = blockIdx.x * 16;
    const int wid  = threadIdx.x >> 5;
    const int lane = threadIdx.x & 31;

    if (wid == 0) {
        tdm_load_2d((unsigned)(uintptr_t)&sAe[0][0], A + (size_t)row0 * FE,
                    16u, (unsigned)FE, (unsigned)FE, 5u /* pad every 64 dw */);
        wait_tensorcnt0();
    }
    __syncthreads();

#pragma unroll
    for (int j = 0; j < 2; ++j) {
        const int col0 = (2 * wid + j) * 16;
        v8f c = {};
        c = gemm_tile_f32<FE>(
            [&](int m, int k) { return sAe[m][k]; },
            [&](int k, int n) { return W[(size_t)(col0 + n) * FE + k]; }, c);
        const int n = lane & 15;
        const float bv = bias[col0 + n];
#pragma unroll
        for (int r = 0; r < 8; ++r) {
            int m = (lane >> 4) * 8 + r;
            float v = c[r] + bv;
            feat_e[(size_t)(row0 + m) * HD + col0 + n] = v;
            sFE[m][col0 + n] = v;
        }
    }
    __syncthreads();
    for (int idx = threadIdx.x; idx < 16 * DD; idx += 256) {
        int i = idx >> 6, d = idx & 63;
        float s = relu_f(sFE[i][d]) + relu_f(sFE[i][DD + d]) +
                  relu_f(sFE[i][2 * DD + d]) + relu_f(sFE[i][3 * DD + d]);
        out_e[(size_t)(row0 + i) * DD + d] = 0.25f * s;
    }
}

// ---------------------------------------------------------------------------
// K3: per (edge, head): leaky-relu attention logits + ordered-int segment max
// ---------------------------------------------------------------------------
__global__ void k_logits_max(const float* __restrict__ feat_src,
                             const float* __restrict__ feat_e,
                             const int* __restrict__ src,
                             const int* __restrict__ dst,
                             const float* __restrict__ attn_n,   // 4 x 64
                             const float* __restrict__ attn_e,   // 4 x 64
                             float* __restrict__ lg1, float* __restrict__ lg2,
                             unsigned* __restrict__ m1, unsigned* __restrict__ m2) {
    const int gw   = blockIdx.x * 8 + (threadIdx.x >> 5);
    const int lane = threadIdx.x & 31;
    const int e = gw >> 2, h = gw & 3;
    const int si = src[e], di = dst[e];
    const size_t bs = (size_t)si * HD + h * DD;
    const size_t bd = (size_t)di * HD + h * DD;
    const size_t be = (size_t)e  * HD + h * DD;
    // stream-prefetch the feat_e rows a few blocks ahead (sequential in e)
    __builtin_prefetch(feat_e + be + (size_t)16 * HD, 0, 0);
    float s1 = 0.0f, s2 = 0.0f;
#pragma unroll
    for (int half = 0; half < 2; ++half) {
        int d = lane + half * 32;
        float fs = feat_src[bs + d];
        float fd = feat_src[bd + d];
        float fe = feat_e[be + d];
        s1 += attn_n[h * DD + d] * (fs + fd);
        s2 += attn_e[h * DD + d] * (fs + fe);
    }
#pragma unroll
    for (int off = 16; off > 0; off >>= 1) {
        s1 += __shfl_xor(s1, off, 32);
        s2 += __shfl_xor(s2, off, 32);
    }
    if (lane == 0) {
        float l1 = s1 > 0.0f ? s1 : 0.2f * s1;
        float l2 = s2 > 0.0f ? s2 : 0.2f * s2;
        lg1[(size_t)e * NH + h] = l1;
        lg2[(size_t)e * NH + h] = l2;
        atomicMax(&m1[(size_t)di * NH + h], f2ord(l1));
        atomicMax(&m2[(size_t)di * NH + h], f2ord(l2));
    }
}

// ---------------------------------------------------------------------------
// K4: ex = exp(logit - max[dst]); atomic denom += ex; atomic num += ex * feat
// ---------------------------------------------------------------------------
__global__ void k_exp_accum(const float* __restrict__ feat_src,
                            const float* __restrict__ feat_e,
                            const int* __restrict__ src,
                            const int* __restrict__ dst,
                            const float* __restrict__ lg1, const float* __restrict__ lg2,
                            const unsigned* __restrict__ m1, const unsigned* __restrict__ m2,
                            float* __restrict__ den1, float* __restrict__ den2,
                            float* __restrict__ num1, float* __restrict__ num2) {
    const int gw   = blockIdx.x * 8 + (threadIdx.x >> 5);
    const int lane = threadIdx.x & 31;
    const int e = gw >> 2, h = gw & 3;
    const int si = src[e], di = dst[e];
    const float ex1 = __expf(lg1[(size_t)e * NH + h] - ord2f(m1[(size_t)di * NH + h]));
    const float ex2 = __expf(lg2[(size_t)e * NH + h] - ord2f(m2[(size_t)di * NH + h]));
    if (lane == 0) {
        atomicAdd(&den1[(size_t)di * NH + h], ex1);
        atomicAdd(&den2[(size_t)di * NH + h], ex2);
    }
    const size_t bs = (size_t)si * HD + h * DD;
    const size_t be = (size_t)e  * HD + h * DD;
    const size_t bn = (size_t)di * HD + h * DD;
    __builtin_prefetch(feat_e + be + (size_t)16 * HD, 0, 0);
#pragma unroll
    for (int half = 0; half < 2; ++half) {
        int d = lane + half * 32;
        atomicAdd(&num1[bn + d], ex1 * feat_src[bs + d]);
        atomicAdd(&num2[bn + d], ex2 * feat_e[be + d]);
    }
}

// ---------------------------------------------------------------------------
// K5: per 16-node block: node_node/edge_node -> W_upd WMMA -> head means ->
//     GRU (gi/gh WMMA) -> gates -> new_node_feats
// ---------------------------------------------------------------------------
__global__ void k_node_final(const float* __restrict__ feat_src,
                             const float* __restrict__ num1, const float* __restrict__ num2,
                             const float* __restrict__ den1, const float* __restrict__ den2,
                             const float* __restrict__ W_upd,   // 64 x 128
                             const float* __restrict__ b_upd,   // 64
                             const float* __restrict__ w_ih,    // 192 x 64
                             const float* __restrict__ w_hh,    // 192 x 64
                             const float* __restrict__ b_ih,    // 192
                             const float* __restrict__ b_hh,    // 192
                             float* __restrict__ out_n) {       // N x 64
    __shared__ float sA[NH][16][2 * DD + PAD];   // concat(node_node, edge_node)
    __shared__ float sH0[16][DD + PAD];          // mean_h feat_src
    __shared__ float sNN[NH][16][DD];            // relu(W_upd @ ...)
    __shared__ float sX[16][DD + PAD];           // mean_h new_node
    __shared__ float sGI[16][3 * DD];            // gi + b_ih
    __shared__ float sGH[16][3 * DD];            // gh + b_hh

    const int i0   = blockIdx.x * 16;
    const int wid  = threadIdx.x >> 5;
    const int lane = threadIdx.x & 31;

    // Stage 1: nn/en staging + h0 means
    for (int idx = threadIdx.x; idx < 16 * HD; idx += 256) {
        int i = idx >> 8, rest = idx & 255;
        int h = rest >> 6, d = rest & 63;
        size_t g = (size_t)(i0 + i) * HD + rest;
        float f  = feat_src[g];
        float d1 = den1[(size_t)(i0 + i) * NH + h];
        float d2 = den2[(size_t)(i0 + i) * NH + h];
        sA[h][i][d]      = relu_f(num1[g] / d1 + f);
        sA[h][i][DD + d] = relu_f(num2[g] / d2 + f);
    }
    for (int idx = threadIdx.x; idx < 16 * DD; idx += 256) {
        int i = idx >> 6, d = idx & 63;
        size_t g = (size_t)(i0 + i) * HD + d;
        sH0[i][d] = 0.25f * (feat_src[g] + feat_src[g + DD] +
                             feat_src[g + 2 * DD] + feat_src[g + 3 * DD]);
    }
    __syncthreads();

    // Stage 2: new_node = relu(concat @ W_upd^T + b), 16 tile-tasks (h, coltile)
#pragma unroll
    for (int j = 0; j < 2; ++j) {
        const int t = wid + j * 8;
        const int h = t >> 2, c0 = (t & 3) * 16;
        v8f c = {};
        c = gemm_tile_f32<2 * DD>(
            [&](int m, int k) { return sA[h][m][k]; },
            [&](int k, int n) { return W_upd[(size_t)(c0 + n) * (2 * DD) + k]; }, c);
        const int n = lane & 15;
        const float bv = b_upd[c0 + n];
#pragma unroll
        for (int r = 0; r < 8; ++r) {
            int m = (lane >> 4) * 8 + r;
            sNN[h][m][c0 + n] = relu_f(c[r] + bv);
        }
    }
    __syncthreads();

    // Stage 3: x = mean over heads
    for (int idx = threadIdx.x; idx < 16 * DD; idx += 256) {
        int i = idx >> 6, d = idx & 63;
        sX[i][d] = 0.25f * (sNN[0][i][d] + sNN[1][i][d] + sNN[2][i][d] + sNN[3][i][d]);
    }
    __syncthreads();

    // Stage 4: gi = x @ w_ih^T + b_ih ; gh = h0 @ w_hh^T + b_hh ; 24 tile-tasks
#pragma unroll
    for (int j = 0; j < 3; ++j) {
        const int t = wid + j * 8;           // 0..23
        const bool is_gi = (t < 12);
        const int c0 = (is_gi ? t : t - 12) * 16;
        v8f c = {};
        if (is_gi) {
            c = gemm_tile_f32<DD>(
                [&](int m, int k) { return sX[m][k]; },
                [&](int k, int n) { return w_ih[(size_t)(c0 + n) * DD + k]; }, c);
        } else {
            c = gemm_tile_f32<DD>(
                [&](int m, int k) { return sH0[m][k]; },
                [&](int k, int n) { return w_hh[(size_t)(c0 + n) * DD + k]; }, c);
        }
        const int n = lane & 15;
        const float bv = is_gi ? b_ih[c0 + n] : b_hh[c0 + n];
#pragma unroll
        for (int r = 0; r < 8; ++r) {
            int m = (lane >> 4) * 8 + r;
            if (is_gi) sGI[m][c0 + n] = c[r] + bv;
            else       sGH[m][c0 + n] = c[r] + bv;
        }
    }
    __syncthreads();

    // Stage 5: GRU gates (PyTorch order r,z,n)
    for (int idx = threadIdx.x; idx < 16 * DD; idx += 256) {
        int i = idx >> 6, d = idx & 63;
        float r  = sigmoid_f(sGI[i][d] + sGH[i][d]);
        float z  = sigmoid_f(sGI[i][DD + d] + sGH[i][DD + d]);
        float ng = tanhf(sGI[i][2 * DD + d] + r * sGH[i][2 * DD + d]);
        float h0 = sH0[i][d];
        out_n[(size_t)(i0 + i) * DD + d] = (1.0f - z) * ng + z * h0;
    }
}

// ---------------------------------------------------------------------------
extern "C" void kernel_launch(void* const* d_in, const int* in_sizes, int n_in,
                              void* d_out, int out_size, void* d_ws, size_t ws_size,
                              hipStream_t stream) {
    const float* node_feats = (const float*)d_in[0];
    const float* edge_feats = (const float*)d_in[1];
    const int*   src        = (const int*)d_in[2];
    const int*   dst        = (const int*)d_in[3];
    const float* W_n2n      = (const float*)d_in[4];
    const float* b_n2n      = (const float*)d_in[5];
    const float* W_e2n      = (const float*)d_in[6];
    const float* b_e2n      = (const float*)d_in[7];
    const float* attn_n2n   = (const float*)d_in[8];
    const float* attn_e2n   = (const float*)d_in[9];
    const float* W_upd      = (const float*)d_in[10];
    const float* b_upd      = (const float*)d_in[11];
    const float* gru_w_ih   = (const float*)d_in[12];
    const float* gru_w_hh   = (const float*)d_in[13];
    const float* gru_b_ih   = (const float*)d_in[14];
    const float* gru_b_hh   = (const float*)d_in[15];

    float* out_n = (float*)d_out;                       // N x 64
    float* out_e = out_n + (size_t)NN * DD;             // E x 64

    // Workspace layout (~371 MB)
    float* ws = (float*)d_ws;
    size_t o = 0;
    float*    feat_src = ws + o; o += (size_t)NN * HD;
    float*    feat_e   = ws + o; o += (size_t)EE * HD;
    float*    num1     = ws + o; o += (size_t)NN * HD;
    float*    num2     = ws + o; o += (size_t)NN * HD;
    float*    den1     = ws + o; o += (size_t)NN * NH;
    float*    den2     = ws + o; o += (size_t)NN * NH;
    unsigned* m1 = (unsigned*)(ws + o); o += (size_t)NN * NH;
    unsigned* m2 = (unsigned*)(ws + o); o += (size_t)NN * NH;
    float*    lg1      = ws + o; o += (size_t)EE * NH;
    float*    lg2      = ws + o; o += (size_t)EE * NH;

    // num1..m2 contiguous; zero (0 bits == 0.0f and ordered-float key floor)
    hipMemsetAsync(num1, 0,
                   ((size_t)2 * NN * HD + (size_t)4 * NN * NH) * sizeof(float),
                   stream);

    k_proj_nodes<<<NN / 16, 256, 0, stream>>>(node_feats, W_n2n, b_n2n, feat_src);
    k_proj_edges<<<EE / 16, 256, 0, stream>>>(edge_feats, W_e2n, b_e2n, feat_e, out_e);
    k_logits_max<<<(EE * NH) / 8, 256, 0, stream>>>(feat_src, feat_e, src, dst,
                                                    attn_n2n, attn_e2n,
                                                    lg1, lg2, m1, m2);
    k_exp_accum<<<(EE * NH) / 8, 256, 0, stream>>>(feat_src, feat_e, src, dst,
                                                   lg1, lg2, m1, m2,
                                                   den1, den2, num1, num2);
    k_node_final<<<NN / 16, 256, 0, stream>>>(feat_src, num1, num2, den1, den2,
                                              W_upd, b_upd, gru_w_ih, gru_w_hh,
                                              gru_b_ih, gru_b_hh, out_n);
}